// VIN_2336462209339
// MI455X (gfx1250) — compile-verified
//
#include <hip/hip_runtime.h>

// VIN on MI455X (gfx1250, wave32).
// Key algebraic optimization: conv(X,Wh)+bh followed by a 1x1 conv (Wr) with no
// nonlinearity collapses to a single 2-channel 3x3 conv with folded weights
// W_eff[2][3][3], b_eff. The remaining 41-step value-iteration stencil runs
// entirely in LDS (one workgroup per batch image), with the fused 20-tap
// conv (Wq on r, w on v) expressed as a chain of 5 V_WMMA_F32_16X16X4_F32.

typedef __attribute__((ext_vector_type(2))) float v2f;
typedef __attribute__((ext_vector_type(8))) float v8f;

#define HH 64
#define WW 64
#define LQ 10
#define RSTR 68          // padded LDS row stride (66 used + 2 slack)
#define HROWS 66         // 64 + 1-pixel zero halo each side
#define NSTEPS 41        // 40 scan steps + initial q0/max folded into same form

// ---------------------------------------------------------------------------
// Kernel 1: fold Wh/bh/Wr into W_eff (18 floats) + b_eff (1 float) in d_ws.
// ---------------------------------------------------------------------------
__global__ __launch_bounds__(32)
void vin_fold_weights(const float* __restrict__ Wh, const float* __restrict__ bh,
                      const float* __restrict__ Wr, float* __restrict__ ws) {
    int l = threadIdx.x;
    if (l < 19) {
        float acc = 0.f;
        for (int c = 0; c < 150; ++c) {
            float s = Wr[c];
            float x = (l < 18) ? Wh[c * 18 + l] : bh[c];
            acc += s * x;
        }
        ws[l] = acc;
    }
}

// ---------------------------------------------------------------------------
// Kernel 2: one workgroup (256 threads = 8 waves) per batch image.
//   r = conv(X, W_eff, pad=1) + b_eff            (LDS, zero halo)
//   41x: t[c] = conv(r,Wq)[c] + conv(v,w)[c]     (5x WMMA f32 16x16x4 per tile)
//        v'  = max_c t[c]                        (in-lane max + shfl_xor 16)
//   last step: write q (t) to global, dot(v', Wc) -> critic.
// ---------------------------------------------------------------------------
__global__ __launch_bounds__(256)
void vin_iter(const float* __restrict__ X, const float* __restrict__ ws,
              const float* __restrict__ Wq, const float* __restrict__ wtr,
              const float* __restrict__ Wc, const float* __restrict__ bc,
              float* __restrict__ out_critic, float* __restrict__ out_q) {
    __shared__ float sR[HROWS * RSTR];
    __shared__ float sV[2][HROWS * RSTR];
    __shared__ float sCrit;

    const int b    = blockIdx.x;
    const int tid  = threadIdx.x;
    const int lane = tid & 31;
    const int wv   = tid >> 5;     // wave id 0..7
    const int lh   = lane >> 4;    // lane half (0: lanes 0-15, 1: lanes 16-31)
    const int n    = lane & 15;    // pixel-in-tile (B/C column), also A row (M)

    // Zero r and both v buffers (halo included); v starts at 0 so step 0
    // produces exactly q0 = conv(r, Wq).
    for (int i = tid; i < HROWS * RSTR; i += 256) {
        sR[i] = 0.f; sV[0][i] = 0.f; sV[1][i] = 0.f;
    }
    if (tid == 0) sCrit = 0.f;
    __syncthreads();

    // ---- r = conv(X, W_eff, pad=1) + b_eff ----
    float weff[18];
    #pragma unroll
    for (int i = 0; i < 18; ++i) weff[i] = ws[i];
    const float beff = ws[18];
    const float* Xb = X + (size_t)b * 2 * HH * WW;
    #pragma unroll 1
    for (int it = 0; it < 16; ++it) {
        int p = tid + 256 * it;           // 0..4095, coalesced
        int y = p >> 6, x = p & 63;
        float acc = beff;
        #pragma unroll
        for (int ci = 0; ci < 2; ++ci)
            #pragma unroll
            for (int t = 0; t < 9; ++t) {
                int yy = y + t / 3 - 1, xx = x + t % 3 - 1;
                if (yy >= 0 && yy < HH && xx >= 0 && xx < WW)
                    acc += weff[ci * 9 + t] * Xb[ci * HH * WW + yy * WW + xx];
            }
        sR[(y + 1) * RSTR + (x + 1)] = acc;
    }
    __syncthreads();

    // ---- A-matrix registers: A[m][k], m = n (channel), k = 4j + g + 2*lh ----
    // k in [0,9): Wq tap k on r; k in [9,18): w tap k-9 on v; k 18/19: zero pad.
    // Zero rows/taps in A make B pad slots don't-care.
    v2f A[5];
    int  off[5][2];    // LDS offset of tap within patch (lane-constant)
    int  isv[5][2];    // 1 -> read from v buffer, 0 -> read from r
    #pragma unroll
    for (int j = 0; j < 5; ++j) {
        float av[2];
        #pragma unroll
        for (int g = 0; g < 2; ++g) {
            int k = 4 * j + g + 2 * lh;
            float wt = 0.f;
            if (n < LQ) {
                if (k < 9)       wt = Wq[n * 9 + k];
                else if (k < 18) wt = wtr[n * 9 + (k - 9)];
            }
            av[g] = wt;
            int kk = (k < 9) ? k : (k - 9);
            if (kk > 8) kk = 8;                       // pad: any valid address
            off[j][g] = (kk / 3) * RSTR + (kk % 3);
            isv[j][g] = (k >= 9);
        }
        A[j].x = av[0]; A[j].y = av[1];
    }

    // ---- 41 fused steps ----
    int cur = 0;
    float critPart = 0.f;
    for (int s = 0; s < NSTEPS; ++s) {
        const bool last = (s == NSTEPS - 1);
        const float* __restrict__ Vc = sV[cur];
        float* __restrict__ Vn = sV[cur ^ 1];
        #pragma unroll 1
        for (int t8 = 0; t8 < 32; ++t8) {
            int tile = wv * 32 + t8;                 // 256 tiles of 16 pixels
            int y  = tile >> 2;
            int x0 = (tile & 3) * 16;
            int baseL = y * RSTR + x0 + n;           // halo coords of patch

            v8f C = {0.f, 0.f, 0.f, 0.f, 0.f, 0.f, 0.f, 0.f};
            #pragma unroll
            for (int j = 0; j < 5; ++j) {
                v2f Bv;
                Bv.x = (isv[j][0] ? Vc : sR)[baseL + off[j][0]];
                Bv.y = (isv[j][1] ? Vc : sR)[baseL + off[j][1]];
                C = __builtin_amdgcn_wmma_f32_16x16x4_f32(
                        /*neg_a=*/false, A[j], /*neg_b=*/false, Bv,
                        /*c_mod=*/(short)0, C, /*reuse_a=*/false, /*reuse_b=*/false);
            }

            // Channel max: lanes 0-15 hold m=0..7 in C[0..7]; lanes 16-31 hold
            // m=8..15 (only m=8,9 valid). Combine halves with shfl_xor(16).
            float pm = lh ? fmaxf(C[0], C[1])
                          : fmaxf(fmaxf(fmaxf(C[0], C[1]), fmaxf(C[2], C[3])),
                                  fmaxf(fmaxf(C[4], C[5]), fmaxf(C[6], C[7])));
            float om   = __shfl_xor(pm, 16, 32);
            float vnew = fmaxf(pm, om);
            if (lh == 0)
                Vn[(y + 1) * RSTR + (x0 + n + 1)] = vnew;

            if (last) {
                int x = x0 + n;
                #pragma unroll
                for (int g = 0; g < 8; ++g) {
                    int mm = g + 8 * lh;
                    if (mm < LQ)
                        out_q[(size_t)b * (LQ * HH * WW) + mm * (HH * WW)
                              + y * WW + x] = C[g];
                }
                if (lh == 0)
                    critPart += vnew * Wc[y * WW + x];
            }
        }
        cur ^= 1;
        __syncthreads();
    }

    // ---- critic reduction (LDS float atomic -> ds_add_f32) ----
    if (lh == 0) atomicAdd(&sCrit, critPart);
    __syncthreads();
    if (tid == 0) out_critic[b] = sCrit + bc[0];
}

// ---------------------------------------------------------------------------
extern "C" void kernel_launch(void* const* d_in, const int* in_sizes, int n_in,
                              void* d_out, int out_size, void* d_ws, size_t ws_size,
                              hipStream_t stream) {
    const float* X  = (const float*)d_in[0];
    const float* Wh = (const float*)d_in[1];
    const float* bh = (const float*)d_in[2];
    const float* Wr = (const float*)d_in[3];
    const float* Wq = (const float*)d_in[4];
    const float* w  = (const float*)d_in[5];
    const float* Wc = (const float*)d_in[6];
    const float* bc = (const float*)d_in[7];

    float* ws         = (float*)d_ws;
    float* out_critic = (float*)d_out;          // [256]
    float* out_q      = (float*)d_out + 256;    // [256, 10*64*64]

    vin_fold_weights<<<1, 32, 0, stream>>>(Wh, bh, Wr, ws);
    vin_iter<<<256, 256, 0, stream>>>(X, ws, Wq, w, Wc, bc, out_critic, out_q);
}